// decoupling_gcn_3075196584280
// MI455X (gfx1250) — compile-verified
//
#include <hip/hip_runtime.h>
#include <hip/hip_bf16.h>

typedef __attribute__((ext_vector_type(2))) float v2f;
typedef __attribute__((ext_vector_type(8))) float v8f;

#define N_    64
#define C_IN  64
#define T_    288
#define V_    19
#define D_    192          // 3 * 64
#define OC    64
#define K_SUB 3
#define G_    8
#define TV    (T_ * V_)    // 5472
#define CNT   ((float)(N_ * T_ * V_))  // 350208
#define BN_EPS 1e-5f

// LDS byte address of a __shared__ object = low 32 bits of its flat address
// (flat LDS addr = {SHARED_BASE, lds_offset} per CDNA5 aperture rules).
__device__ __forceinline__ unsigned lds_addr(const void* p) {
    return (unsigned)(unsigned long long)p;
}

// Async global->LDS DMA (16B per lane), tracked by ASYNCcnt.
__device__ __forceinline__ void async_copy_b128(unsigned lds, unsigned gbyteoff,
                                                const void* base) {
    asm volatile("global_load_async_to_lds_b128 %0, %1, %2"
                 :: "v"(lds), "v"(gbyteoff),
                    "s"((unsigned long long)base)
                 : "memory");
}
__device__ __forceinline__ void wait_async0() {
    asm volatile("s_wait_asynccnt 0x0" ::: "memory");
}

// ---------------------------------------------------------------------------
// 1) Normalize adjacency: normA[k,g,v,w] = A[k,g,v,w] / (sqrt(8*sum_g A^2)+1e-4)
// ---------------------------------------------------------------------------
__global__ void k_norm_adj(const float* __restrict__ A, float* __restrict__ normA) {
    int idx = blockIdx.x * blockDim.x + threadIdx.x;
    if (idx >= K_SUB * G_ * V_ * V_) return;
    int w = idx % V_;
    int v = (idx / V_) % V_;
    int k = idx / (G_ * V_ * V_);
    float s = 0.f;
    for (int g = 0; g < G_; ++g) {
        float a = A[((k * G_ + g) * V_ + v) * V_ + w];
        s += a * a;
    }
    float nrm = sqrtf(8.0f * s) + 1e-4f;
    normA[idx] = A[idx] / nrm;
}

// ---------------------------------------------------------------------------
// 2) GEMM1: y[n,d,tv] = sum_c x[n,c,tv] * W[c,d] + bias[d], fused bn0 stats.
//    f32 WMMA 16x16x4; x tile staged via async global->LDS B128 DMA.
// ---------------------------------------------------------------------------
__global__ __launch_bounds__(384) void k_gemm1(
    const float* __restrict__ x, const float* __restrict__ W,
    const float* __restrict__ bias, float* __restrict__ y,
    float* __restrict__ stats0) {
    __shared__ __align__(16) float xs[C_IN * 16];   // 64 channels x 16 tv (4 KB)

    const int n   = blockIdx.y;
    const int tv0 = blockIdx.x * 16;
    const int tid = threadIdx.x;

    // --- async DMA: stage x tile, 256 lanes x 16B, no VGPR round-trip ---
    if (tid < 256) {
        const int c  = tid >> 2;       // channel row
        const int ch = tid & 3;        // 4-float chunk within the 16-wide tile
        const unsigned dst = lds_addr(&xs[c * 16 + ch * 4]);
        const unsigned src = (unsigned)(((n * C_IN + c) * TV + tv0 + ch * 4) * 4);
        async_copy_b128(dst, src, x);
    }

    const int wave = tid >> 5;
    const int lane = tid & 31;
    const int d0   = wave * 16;
    const int m    = lane & 15;   // row (A) / col (B) within tile
    const int hi   = lane >> 4;   // lane half

    // --- overlap: load W A-fragments into registers while DMA runs ---
    v2f a[16];
#pragma unroll
    for (int s = 0; s < 16; ++s) {
#pragma unroll
        for (int j = 0; j < 2; ++j) {
            const int kkA = hi * 2 + j;                 // A frag K index
            a[s][j] = W[(4 * s + kkA) * D_ + d0 + m];
        }
    }

    wait_async0();
    __syncthreads();

    // --- B fragments from LDS ---
    v2f b[16];
#pragma unroll
    for (int s = 0; s < 16; ++s) {
#pragma unroll
        for (int j = 0; j < 2; ++j) {
            const int kkB = j * 2 + hi;                 // B frag K index
            b[s][j] = xs[(4 * s + kkB) * 16 + m];
        }
    }

    v8f acc = {0.f, 0.f, 0.f, 0.f, 0.f, 0.f, 0.f, 0.f};
#pragma unroll
    for (int s = 0; s < 16; ++s) {
        acc = __builtin_amdgcn_wmma_f32_16x16x4_f32(
            false, a[s], false, b[s], (short)0, acc, false, false);
    }

    // epilogue: +bias, store y, fused per-d sum / sumsq for bn0
#pragma unroll
    for (int j = 0; j < 8; ++j) {
        const int d = d0 + hi * 8 + j;                  // C layout: vgpr j -> M=j (+8 hi)
        float val = acc[j] + bias[d];
        y[((long)(n * D_ + d)) * TV + tv0 + m] = val;

        float s1 = val, s2 = val * val;
        for (int dlt = 1; dlt < 16; dlt <<= 1) {
            s1 += __shfl_xor(s1, dlt, 16);
            s2 += __shfl_xor(s2, dlt, 16);
        }
        if (m == 0) {
            atomicAdd(&stats0[d], s1);
            atomicAdd(&stats0[D_ + d], s2);
        }
    }
}

// ---------------------------------------------------------------------------
// 3a) Fold bn0 scale into adjacency: A2[c][k][v][w] = rs(d) * normA[k, c%8, v, w]
// ---------------------------------------------------------------------------
__global__ void k_prep_a2(const float* __restrict__ normA,
                          const float* __restrict__ stats0,
                          float* __restrict__ A2) {
    int idx = blockIdx.x * blockDim.x + threadIdx.x;
    if (idx >= OC * K_SUB * V_ * V_) return;
    int w = idx % V_;
    int v = (idx / V_) % V_;
    int k = (idx / (V_ * V_)) % K_SUB;
    int c = idx / (K_SUB * V_ * V_);
    int d = k * OC + c;
    float mean = stats0[d] / CNT;
    float var  = stats0[D_ + d] / CNT - mean * mean;
    float rs   = rsqrtf(var + BN_EPS);
    A2[idx] = rs * normA[((k * G_ + (c & 7)) * V_ + v) * V_ + w];
}

// 3b) Fold bn0 mean into per-(c,w) bias: bias2[c,w] = sum_k (-m*rs) * sum_v normA
__global__ void k_bias2(const float* __restrict__ normA,
                        const float* __restrict__ stats0,
                        float* __restrict__ bias2) {
    int idx = blockIdx.x * blockDim.x + threadIdx.x;
    if (idx >= OC * V_) return;
    int w = idx % V_;
    int c = idx / V_;
    float b = 0.f;
    for (int k = 0; k < K_SUB; ++k) {
        int d = k * OC + c;
        float mean = stats0[d] / CNT;
        float var  = stats0[D_ + d] / CNT - mean * mean;
        float rs   = rsqrtf(var + BN_EPS);
        float sA = 0.f;
        for (int v = 0; v < V_; ++v)
            sA += normA[((k * G_ + (c & 7)) * V_ + v) * V_ + w];
        b += (-mean * rs) * sA;
    }
    bias2[idx] = b;
}

// ---------------------------------------------------------------------------
// 4) GEMM2: z[n,c,t,w] = sum_{k,v} y[n,k*64+c,t,v] * A2[c,k,v,w] + bias2[c,w]
//    y tile staged via async global->LDS B128 DMA; fused per-c BN stats.
// ---------------------------------------------------------------------------
__global__ __launch_bounds__(640) void k_gemm2(
    const float* __restrict__ y, const float* __restrict__ A2,
    const float* __restrict__ bias2, float* __restrict__ z,
    float* __restrict__ stats1) {
    __shared__ __align__(16) float As[K_SUB * V_ * V_];   // 1083 floats
    __shared__ __align__(16) float ys[K_SUB * 32 * V_];   // 1824 floats
    __shared__ float red[2 * 20];

    const int c   = blockIdx.y;
    const int n   = blockIdx.z;
    const int t0  = blockIdx.x * 32;
    const int tid = threadIdx.x;

    // --- async DMA: 3 rows x 608 floats = 456 lanes x 16B ---
    if (tid < 456) {
        const int k = tid / 152;
        const int i = tid % 152;
        const unsigned dst = lds_addr(&ys[k * (32 * V_) + i * 4]);
        const unsigned src =
            (unsigned)((((long)(n * D_ + k * OC + c)) * TV + t0 * V_ + i * 4) * 4);
        async_copy_b128(dst, src, y);
    }
    // overlap: stage A2 tile through the normal path
    for (int i = tid; i < K_SUB * V_ * V_; i += 640)
        As[i] = A2[c * (K_SUB * V_ * V_) + i];

    wait_async0();
    __syncthreads();

    const bool active = tid < 32 * V_;
    float zval = 0.f;
    if (active) {
        const int tt = tid / V_;
        const int w  = tid % V_;
        zval = bias2[c * V_ + w];
#pragma unroll
        for (int k = 0; k < K_SUB; ++k)
#pragma unroll
            for (int v = 0; v < V_; ++v)
                zval += ys[(k * 32 + tt) * V_ + v] * As[k * (V_ * V_) + v * V_ + w];
        z[((long)(n * OC + c)) * TV + (t0 + tt) * V_ + w] = zval;
    }

    float s1 = active ? zval : 0.f;
    float s2 = active ? zval * zval : 0.f;
    for (int dlt = 16; dlt; dlt >>= 1) {
        s1 += __shfl_xor(s1, dlt, 32);
        s2 += __shfl_xor(s2, dlt, 32);
    }
    const int wv = tid >> 5, ln = tid & 31;
    if (ln == 0) { red[wv] = s1; red[20 + wv] = s2; }
    __syncthreads();
    if (tid == 0) {
        float S = 0.f, Q = 0.f;
        for (int i = 0; i < 20; ++i) { S += red[i]; Q += red[20 + i]; }
        atomicAdd(&stats1[c], S);
        atomicAdd(&stats1[OC + c], Q);
    }
}

// 5) finalize second-BN stats -> mean, rsqrt per c
__global__ void k_fin1(const float* __restrict__ stats1, float* __restrict__ ms) {
    int c = threadIdx.x;
    if (c >= OC) return;
    float mean = stats1[c] / CNT;
    float var  = stats1[OC + c] / CNT - mean * mean;
    ms[c]      = mean;
    ms[OC + c] = rsqrtf(var + BN_EPS);
}

// 6) out = relu(1e-6 * (z - m) * rs + x)   (float4 streaming; TV % 4 == 0)
__global__ void k_out(const float* __restrict__ z, const float* __restrict__ x,
                      const float* __restrict__ ms, float* __restrict__ out) {
    long i4 = (long)blockIdx.x * blockDim.x + threadIdx.x;
    const long total4 = (long)N_ * OC * TV / 4;
    if (i4 >= total4) return;
    int c = (int)((i4 * 4 / TV) % OC);
    float4 zv = ((const float4*)z)[i4];
    float4 xv = ((const float4*)x)[i4];
    float m = ms[c], rs = ms[OC + c];
    const float g = 1e-6f;
    float4 o;
    o.x = fmaxf(g * (zv.x - m) * rs + xv.x, 0.f);
    o.y = fmaxf(g * (zv.y - m) * rs + xv.y, 0.f);
    o.z = fmaxf(g * (zv.z - m) * rs + xv.z, 0.f);
    o.w = fmaxf(g * (zv.w - m) * rs + xv.w, 0.f);
    ((float4*)out)[i4] = o;
}

// ---------------------------------------------------------------------------
extern "C" void kernel_launch(void* const* d_in, const int* in_sizes, int n_in,
                              void* d_out, int out_size, void* d_ws, size_t ws_size,
                              hipStream_t stream) {
    const float* x  = (const float*)d_in[0];   // [64,64,288,19]
    const float* A  = (const float*)d_in[1];   // [3,8,19,19]
    const float* W  = (const float*)d_in[2];   // [64,192]
    const float* Lb = (const float*)d_in[3];   // [1,192,1,1]
    float* out = (float*)d_out;

    float* ws = (float*)d_ws;
    size_t off = 0;
    float* y      = ws + off; off += (size_t)N_ * D_ * TV;       // 67,239,936
    float* z      = ws + off; off += (size_t)N_ * OC * TV;       // 22,413,312
    float* stats0 = ws + off; off += 2 * D_;                     // 384
    float* stats1 = ws + off; off += 2 * OC;                     // 128
    float* ms1    = ws + off; off += 2 * OC;                     // 128
    float* normA  = ws + off; off += K_SUB * G_ * V_ * V_;       // 8,664
    float* A2     = ws + off; off += OC * K_SUB * V_ * V_;       // 69,312
    float* bias2  = ws + off; off += OC * V_;                    // 1,216

    // zero the atomic-accumulated stats every call (graph-capturable memset)
    hipMemsetAsync(stats0, 0, (2 * D_ + 2 * OC) * sizeof(float), stream);

    k_norm_adj<<<(K_SUB * G_ * V_ * V_ + 255) / 256, 256, 0, stream>>>(A, normA);

    dim3 g1(TV / 16, N_);                       // 342 x 64
    k_gemm1<<<g1, 384, 0, stream>>>(x, W, Lb, y, stats0);

    k_prep_a2<<<(OC * K_SUB * V_ * V_ + 255) / 256, 256, 0, stream>>>(normA, stats0, A2);
    k_bias2<<<(OC * V_ + 255) / 256, 256, 0, stream>>>(normA, stats0, bias2);

    dim3 g2(T_ / 32, OC, N_);                   // 9 x 64 x 64
    k_gemm2<<<g2, 640, 0, stream>>>(y, A2, bias2, z, stats1);

    k_fin1<<<1, 64, 0, stream>>>(stats1, ms1);

    const long total4 = (long)N_ * OC * TV / 4;
    k_out<<<(int)((total4 + 255) / 256), 256, 0, stream>>>(z, x, ms1, out);
}